// Qwen3Attention_57586921504833
// MI455X (gfx1250) — compile-verified
//
#include <hip/hip_runtime.h>
#include <hip/hip_bf16.h>

// Problem constants (match reference)
#define B_   2
#define S_   1024
#define H_   4096
#define NH_  32
#define NKV_ 8
#define D_   128
#define EPS_ 1e-6f
#define INV_SQRT_D 0.08838834764831845f  // 1/sqrt(128)

typedef __attribute__((ext_vector_type(2))) float v2f;
typedef __attribute__((ext_vector_type(8))) float v8f;

static __device__ __forceinline__ v8f wmma_f32(v2f a, v2f b, v8f c) {
  // D = A(16x4) * B(4x16) + C(16x16), full fp32 matrix path on CDNA5
  return __builtin_amdgcn_wmma_f32_16x16x4_f32(
      /*neg_a=*/false, a, /*neg_b=*/false, b,
      /*c_mod=*/(short)0, c, /*reuse_a=*/false, /*reuse_b=*/false);
}

// -----------------------------------------------------------------------------
// Y(M,N) = X(M,K) * W(N,K)^T
// Each wave computes a 16(M) x 64(N) strip (4 accumulators). Per k-chunk of 16:
// issue all 20 fragment loads first (4 A + 16 B, independent base pointers with
// immediate offsets -> load clauses + staggered s_wait_loadcnt), then 16 WMMAs.
// A-fragment: lane(0..15) holds row M=lane; v0/v1 = K=2*half / +1 (half=lane>>4).
// B-fragment: lane(0..15) ~ column N; halves ~ K pairs (mirrors A layout).
// -----------------------------------------------------------------------------
__global__ __launch_bounds__(128) void gemm_xwt(
    const float* __restrict__ X, const float* __restrict__ W,
    float* __restrict__ Y, int M, int N, int K)
{
  const int lane = threadIdx.x & 31;
  const int wave = threadIdx.x >> 5;
  const int lrow = lane & 15;
  const int half = lane >> 4;
  const int tn0  = blockIdx.x * 64;         // N strip base (4 tiles of 16)
  const int tm   = blockIdx.y * 4 + wave;   // M tile index

  const float* xp  = X + (size_t)(tm * 16 + lrow) * K + 2 * half;
  const float* wp0 = W + (size_t)(tn0 + lrow) * K + 2 * half;
  const float* wp1 = wp0 + (size_t)16 * K;  // independent induction variables
  const float* wp2 = wp1 + (size_t)16 * K;
  const float* wp3 = wp2 + (size_t)16 * K;

  v8f acc0 = {}, acc1 = {}, acc2 = {}, acc3 = {};

  for (int k = 0; k < K; k += 16) {
    v2f a[4], b0[4], b1[4], b2[4], b3[4];
    // ---- issue all loads for this chunk (clause-friendly, immediate offsets)
    #pragma unroll
    for (int kk = 0; kk < 4; ++kk) {
      const int o = k + 4 * kk;
      a[kk]  = *(const v2f*)(xp  + o);
      b0[kk] = *(const v2f*)(wp0 + o);
      b1[kk] = *(const v2f*)(wp1 + o);
      b2[kk] = *(const v2f*)(wp2 + o);
      b3[kk] = *(const v2f*)(wp3 + o);
    }
    // ---- then consume: 16 WMMAs, waits can stagger as loads drain
    #pragma unroll
    for (int kk = 0; kk < 4; ++kk) {
      acc0 = wmma_f32(a[kk], b0[kk], acc0);
      acc1 = wmma_f32(a[kk], b1[kk], acc1);
      acc2 = wmma_f32(a[kk], b2[kk], acc2);
      acc3 = wmma_f32(a[kk], b3[kk], acc3);
    }
  }

  // C layout: lanes 0-15 hold rows 0..7 (vgpr r -> row r), lanes 16-31 rows 8..15
  const int row0 = tm * 16 + half * 8;
  float* yp = Y + (size_t)row0 * N + tn0 + lrow;
  #pragma unroll
  for (int r = 0; r < 8; ++r) {
    yp[(size_t)r * N +  0] = acc0[r];
    yp[(size_t)r * N + 16] = acc1[r];
    yp[(size_t)r * N + 32] = acc2[r];
    yp[(size_t)r * N + 48] = acc3[r];
  }
}

// -----------------------------------------------------------------------------
// Fused RMSNorm (over D=128) + RoPE. One wave per (b,s,head) row.
// Lane l holds d = 4l..4l+3; partner element d±64 is exactly 16 lanes away.
// Output layout: (B, HN, S, D).
// -----------------------------------------------------------------------------
__global__ __launch_bounds__(256) void rmsnorm_rope(
    const float* __restrict__ P,    // (B*S, HN*D) projection output
    const float* __restrict__ cs,   // (B*S, D)
    const float* __restrict__ sn,   // (B*S, D)
    const float* __restrict__ nw,   // (D)
    float* __restrict__ O,          // (B, HN, S, D)
    int HN, int nrows)
{
  const int wid = blockIdx.x * 8 + (threadIdx.x >> 5);
  if (wid >= nrows) return;
  const int lane = threadIdx.x & 31;
  const int hd = wid % HN;
  const int bs = wid / HN;
  const int s  = bs % S_;
  const int b  = bs / S_;
  const int d  = lane * 4;

  float4 x = *(const float4*)(P + (size_t)bs * HN * D_ + (size_t)hd * D_ + d);

  float ss = x.x * x.x + x.y * x.y + x.z * x.z + x.w * x.w;
  #pragma unroll
  for (int off = 16; off; off >>= 1) ss += __shfl_xor(ss, off);
  const float inv = rsqrtf(ss * (1.0f / D_) + EPS_);

  float4 w4 = *(const float4*)(nw + d);
  x.x *= inv * w4.x; x.y *= inv * w4.y; x.z *= inv * w4.z; x.w *= inv * w4.w;

  // rotate_half via cross-half shuffle: lanes 0-15 see x[d+64], lanes 16-31 x[d-64]
  float4 oth;
  oth.x = __shfl_xor(x.x, 16);
  oth.y = __shfl_xor(x.y, 16);
  oth.z = __shfl_xor(x.z, 16);
  oth.w = __shfl_xor(x.w, 16);
  const float sgn = (lane < 16) ? -1.0f : 1.0f;

  float4 c4 = *(const float4*)(cs + (size_t)bs * D_ + d);
  float4 s4 = *(const float4*)(sn + (size_t)bs * D_ + d);
  float4 o;
  o.x = x.x * c4.x + sgn * oth.x * s4.x;
  o.y = x.y * c4.y + sgn * oth.y * s4.y;
  o.z = x.z * c4.z + sgn * oth.z * s4.z;
  o.w = x.w * c4.w + sgn * oth.w * s4.w;

  *(float4*)(O + (((size_t)b * HN + hd) * S_ + s) * D_ + d) = o;
}

// -----------------------------------------------------------------------------
// Causal GQA flash attention. 8 waves/block; each wave owns a 16-row Q strip
// (block covers 128 q rows of one (b,h)). Q fragments pre-staged in registers.
// Per 16-key tile: QK^T (32 wmma), online softmax, P via LDS transpose, PV
// (32 wmma) into a 16x128 fp32 accumulator. Next K/V tiles prefetched.
// -----------------------------------------------------------------------------
__global__ __launch_bounds__(256) void attn_kernel(
    const float* __restrict__ Qr,   // (B, NH, S, D)
    const float* __restrict__ Kr,   // (B, NKV, S, D)
    const float* __restrict__ Vp,   // (B*S, NKV*D)
    float* __restrict__ Oa)         // (B*S, NH*D)
{
  __shared__ float plds[8][16][17];  // per-wave P tile, padded vs bank conflicts

  const int w    = threadIdx.x >> 5;
  const int lane = threadIdx.x & 31;
  const int lrow = lane & 15;
  const int half = lane >> 4;

  const int b  = blockIdx.z;
  const int h  = blockIdx.y;
  const int kv = h / (NH_ / NKV_);
  const int qi0 = blockIdx.x * 128 + w * 16;

  const float* Qh = Qr + ((size_t)b * NH_ + h) * S_ * D_;
  const float* Kh = Kr + ((size_t)b * NKV_ + kv) * S_ * D_;
  const float* Vh = Vp + (size_t)b * S_ * NKV_ * D_ + (size_t)kv * D_;

  // Pre-stage Q strip as 32 A-fragments (16x128 -> 32 chunks of K=4)
  v2f aq[32];
  {
    const float* qrow = Qh + (size_t)(qi0 + lrow) * D_ + 2 * half;
    #pragma unroll
    for (int c = 0; c < 32; ++c) aq[c] = *(const v2f*)(qrow + 4 * c);
  }

  v8f o[8];
  float m[8], l[8];
  #pragma unroll
  for (int t = 0; t < 8; ++t) o[t] = (v8f){};
  #pragma unroll
  for (int r = 0; r < 8; ++r) { m[r] = -1e30f; l[r] = 0.0f; }

  const int ktmax = (qi0 + 15) >> 4;
  for (int kt = 0; kt <= ktmax; ++kt) {
    const int kbase = kt << 4;

    // Prefetch next tile's K row and V row for this lane (global_prefetch_b8)
    if (kt < ktmax) {
      __builtin_prefetch(Kh + (size_t)(kbase + 16 + lrow) * D_, 0, 1);
      __builtin_prefetch(Vh + (size_t)(kbase + 16 + lrow) * (NKV_ * D_), 0, 1);
    }

    // ---- S = Q K^T (16x16 tile): batch 8 B-fragment loads, then 8 WMMAs ----
    const float* krow = Kh + (size_t)(kbase + lrow) * D_ + 2 * half;
    v8f sc = (v8f){};
    #pragma unroll
    for (int g = 0; g < 4; ++g) {
      v2f bf[8];
      #pragma unroll
      for (int c = 0; c < 8; ++c)
        bf[c] = *(const v2f*)(krow + 4 * (g * 8 + c));
      #pragma unroll
      for (int c = 0; c < 8; ++c)
        sc = wmma_f32(aq[g * 8 + c], bf[c], sc);
    }

    // ---- online softmax (per-lane: col j fixed, rows r+8*half) ----
    const int j = kbase + lrow;
    float p[8];
    #pragma unroll
    for (int r = 0; r < 8; ++r) {
      const int qrow_g = qi0 + r + 8 * half;
      float sv = sc[r] * INV_SQRT_D;
      if (j > qrow_g) sv = -1e30f;   // causal mask
      float rmax = sv;
      #pragma unroll
      for (int off = 1; off < 16; off <<= 1)
        rmax = fmaxf(rmax, __shfl_xor(rmax, off));
      const float mn    = fmaxf(m[r], rmax);
      const float scale = __expf(m[r] - mn);
      const float pv    = __expf(sv - mn);
      float rs = pv;
      #pragma unroll
      for (int off = 1; off < 16; off <<= 1) rs += __shfl_xor(rs, off);
      l[r] = l[r] * scale + rs;
      m[r] = mn;
      #pragma unroll
      for (int t = 0; t < 8; ++t) o[t][r] *= scale;
      p[r] = pv;
    }

    // ---- transpose P (C layout -> A layout) through per-wave LDS ----
    #pragma unroll
    for (int r = 0; r < 8; ++r) plds[w][r + 8 * half][lrow] = p[r];
    __builtin_amdgcn_wave_barrier();
    asm volatile("s_wait_dscnt 0" ::: "memory");  // cross-lane LDS visibility
    v2f ap[4];
    #pragma unroll
    for (int c2 = 0; c2 < 4; ++c2) {
      ap[c2].x = plds[w][lrow][4 * c2 + 2 * half];
      ap[c2].y = plds[w][lrow][4 * c2 + 2 * half + 1];
    }
    __builtin_amdgcn_wave_barrier();

    // ---- O += P * V (16x16 * 16x128): per 16-col tile, load 4 B frags then
    //      4 WMMAs ----
    const float* vbase = Vh + (size_t)(kbase + 2 * half) * (NKV_ * D_) + lrow;
    #pragma unroll
    for (int t = 0; t < 8; ++t) {
      v2f bv[4];
      #pragma unroll
      for (int c2 = 0; c2 < 4; ++c2) {
        bv[c2].x = vbase[(size_t)(4 * c2) * (NKV_ * D_) + t * 16];
        bv[c2].y = vbase[(size_t)(4 * c2 + 1) * (NKV_ * D_) + t * 16];
      }
      v8f acc = o[t];
      #pragma unroll
      for (int c2 = 0; c2 < 4; ++c2) acc = wmma_f32(ap[c2], bv[c2], acc);
      o[t] = acc;
    }
  }

  // ---- normalize and store to (B*S, NH*D) ----
  #pragma unroll
  for (int r = 0; r < 8; ++r) {
    const float invl = 1.0f / l[r];
    const int row_g = qi0 + r + 8 * half;
    float* dst = Oa + ((size_t)b * S_ + row_g) * (NH_ * D_) + (size_t)h * D_;
    #pragma unroll
    for (int t = 0; t < 8; ++t) dst[t * 16 + lrow] = o[t][r] * invl;
  }
}

// -----------------------------------------------------------------------------
// Host launcher
// -----------------------------------------------------------------------------
extern "C" void kernel_launch(void* const* d_in, const int* in_sizes, int n_in,
                              void* d_out, int out_size, void* d_ws, size_t ws_size,
                              hipStream_t stream) {
  const float* hidden = (const float*)d_in[0];   // (B,S,H)
  const float* cosb   = (const float*)d_in[1];   // (B,S,D)
  const float* sinb   = (const float*)d_in[2];   // (B,S,D)
  const float* Wq     = (const float*)d_in[3];   // (NH*D, H)
  const float* Wk     = (const float*)d_in[4];   // (NKV*D, H)
  const float* Wv     = (const float*)d_in[5];   // (NKV*D, H)
  const float* Wo     = (const float*)d_in[6];   // (H, NH*D)
  const float* qnw    = (const float*)d_in[7];   // (D)
  const float* knw    = (const float*)d_in[8];   // (D)
  float* out = (float*)d_out;

  const size_t nQ = (size_t)B_ * S_ * NH_ * D_;   // 8388608
  const size_t nK = (size_t)B_ * S_ * NKV_ * D_;  // 2097152

  float* ws     = (float*)d_ws;
  float* q_proj = ws;                 // also reused as attention output
  float* k_proj = q_proj + nQ;
  float* v_proj = k_proj + nK;
  float* q_rot  = v_proj + nK;
  float* k_rot  = q_rot + nQ;

  const int M = B_ * S_;  // 2048

  // QKV projections (Y = X * W^T); grid = (N/64, M/64), 4 waves/block
  gemm_xwt<<<dim3((NH_ * D_) / 64, M / 64), 128, 0, stream>>>(
      hidden, Wq, q_proj, M, NH_ * D_, H_);
  gemm_xwt<<<dim3((NKV_ * D_) / 64, M / 64), 128, 0, stream>>>(
      hidden, Wk, k_proj, M, NKV_ * D_, H_);
  gemm_xwt<<<dim3((NKV_ * D_) / 64, M / 64), 128, 0, stream>>>(
      hidden, Wv, v_proj, M, NKV_ * D_, H_);

  // RMSNorm + RoPE, re-layout to (B, heads, S, D)
  rmsnorm_rope<<<(M * NH_) / 8, 256, 0, stream>>>(
      q_proj, cosb, sinb, qnw, q_rot, NH_, M * NH_);
  rmsnorm_rope<<<(M * NKV_) / 8, 256, 0, stream>>>(
      k_proj, cosb, sinb, knw, k_rot, NKV_, M * NKV_);

  // Flash attention (writes into q_proj region, which is free now)
  attn_kernel<<<dim3(S_ / 128, NH_, B_), 256, 0, stream>>>(
      q_rot, k_rot, v_proj, q_proj);

  // Output projection
  gemm_xwt<<<dim3(H_ / 64, M / 64), 128, 0, stream>>>(
      q_proj, Wo, out, M, H_, NH_ * D_);
}